// ChannelDeconv_31610959298817
// MI455X (gfx1250) — compile-verified
//
#include <hip/hip_runtime.h>

typedef __attribute__((ext_vector_type(2))) float v2f;
typedef __attribute__((ext_vector_type(8))) float v8f;

#define NCH    64                    // BLOCK / channel count
#define NIMG   32
#define HWSZ   65536                 // 256*256
#define TOTJ   (NIMG * HWSZ)         // 2097152 columns of x1
#define S2     9                     // STRIDE^2
#define NSUB   ((TOTJ + S2 - 1) / S2)   // 233017 subsampled columns
#define NCHUNK ((NSUB + 63) / 64)       // 3641 chunks of 64 columns
#define EPSV   0.01f

// ---------------------------------------------------------------------------
// Stage one 64x64 subsampled tile into LDS.
// Full chunks: GLOBAL_LOAD_ASYNC_TO_LDS_B32 (ASYNCcnt-tracked, no VGPR hop).
// Tail chunk: guarded scalar loads + ds stores (zero-padded -> contributes 0).
// LDS byte offset = low 32 bits of the generic pointer (flat-LDS aperture:
// see ISA 10.2).
// ---------------------------------------------------------------------------
__device__ __forceinline__
void k1_issue_tile(const float* __restrict__ x, float* __restrict__ dst,
                   int chunk, int tid) {
  const bool full = (chunk * 64 + 63) < NSUB;
  if (full) {
    #pragma unroll
    for (int s = 0; s < 8; ++s) {
      int i  = tid + s * 512;
      int c  = i >> 6;
      int tt = i & 63;
      int j  = (chunk * 64 + tt) * S2;
      const float* gp = x + (((j >> 16) * NCH + c) << 16) + (j & (HWSZ - 1));
      unsigned ldsoff = (unsigned)(size_t)(dst + c * 65 + tt);
      asm volatile("global_load_async_to_lds_b32 %0, %1, off"
                   :: "v"(ldsoff), "v"(gp) : "memory");
    }
  } else {
    #pragma unroll
    for (int s = 0; s < 8; ++s) {
      int   i  = tid + s * 512;
      int   c  = i >> 6;
      int   tt = i & 63;
      int   t  = chunk * 64 + tt;
      float v  = 0.f;
      if (t < NSUB) {
        int j = t * S2;
        v = x[(((j >> 16) * NCH + c) << 16) + (j & (HWSZ - 1))];
      }
      dst[c * 65 + tt] = v;
    }
  }
}

// ---------------------------------------------------------------------------
// k1: per-block partial cov (Xs Xs^T) and column-sum (for mean), via WMMA f32.
// 512 threads = 16 waves; wave w owns cov tile (ci=w>>2, cj=w&3).
// Double-buffered LDS tile (64ch x 64col, stride 65 -> conflict-free frags);
// next chunk's async loads overlap current chunk's WMMA.
// ---------------------------------------------------------------------------
__global__ void __launch_bounds__(512)
k1_cov_partial(const float* __restrict__ x, float* __restrict__ covPart,
               float* __restrict__ meanPart) {
  __shared__ float sT[2][NCH * 65];
  const int tid  = threadIdx.x;
  const int lane = tid & 31;
  const int wave = tid >> 5;
  const int ci   = wave >> 2;
  const int cj   = wave & 3;
  const int mcol = lane & 15;
  const int koff = (lane < 16) ? 0 : 2;

  v8f   acc  = {};
  float msum = 0.f;

  int buf   = 0;
  int chunk = blockIdx.x;
  if (chunk < NCHUNK) k1_issue_tile(x, sT[0], chunk, tid);

  for (; chunk < NCHUNK; chunk += gridDim.x) {
    asm volatile("s_wait_asynccnt 0" ::: "memory");   // own async slice landed
    __syncthreads();                                  // whole tile[buf] ready

    const int next = chunk + gridDim.x;
    if (next < NCHUNK) k1_issue_tile(x, sT[buf ^ 1], next, tid);

    const float* tb = sT[buf];
    if (tid < NCH) {                                  // waves 0,1: uniform
      #pragma unroll
      for (int tt = 0; tt < 64; ++tt) msum += tb[tid * 65 + tt];
    }

    #pragma unroll
    for (int kc = 0; kc < 16; ++kc) {
      v2f a, b;
      // A = Xs[ci*16 + m, k-chunk]  (16x4, lane m = L%16, VGPR0->K koff)
      a[0] = tb[(ci * 16 + mcol) * 65 + kc * 4 + koff];
      a[1] = tb[(ci * 16 + mcol) * 65 + kc * 4 + koff + 1];
      // B = Xs^T[k-chunk, cj*16 + n] = Xs[cj*16 + n, k-chunk]
      b[0] = tb[(cj * 16 + mcol) * 65 + kc * 4 + koff];
      b[1] = tb[(cj * 16 + mcol) * 65 + kc * 4 + koff + 1];
      acc = __builtin_amdgcn_wmma_f32_16x16x4_f32(false, a, false, b,
                                                  (short)0, acc, false, false);
    }
    __syncthreads();                                  // done reading tile[buf]
    buf ^= 1;
  }

  // write partial 16x16 tile (C/D layout: VGPR r -> row r (+8 for hi half))
  float* cp = covPart + (size_t)blockIdx.x * (NCH * NCH);
  #pragma unroll
  for (int r = 0; r < 8; ++r) {
    int row = ci * 16 + r + ((lane < 16) ? 0 : 8);
    int col = cj * 16 + mcol;
    cp[row * NCH + col] = acc[r];
  }
  if (tid < NCH) meanPart[blockIdx.x * NCH + tid] = msum;
}

// ---------------------------------------------------------------------------
// k2: fixed-order reduction of partials -> cov = S/n + eps*I, mean = sum/n.
// ---------------------------------------------------------------------------
__global__ void __launch_bounds__(256)
k2_finalize(const float* __restrict__ covPart, const float* __restrict__ meanPart,
            float* __restrict__ cov, float* __restrict__ meanv, int nb) {
  const int   tid = threadIdx.x;
  const float inv = 1.0f / (float)NSUB;
  for (int i = tid; i < NCH * NCH; i += 256) {
    float s = 0.f;
    for (int b = 0; b < nb; ++b) s += covPart[(size_t)b * (NCH * NCH) + i];
    int r = i >> 6, c = i & 63;
    cov[i] = s * inv + ((r == c) ? EPSV : 0.f);
  }
  if (tid < NCH) {
    float s = 0.f;
    for (int b = 0; b < nb; ++b) s += meanPart[b * NCH + tid];
    meanv[tid] = s * inv;
  }
}

// ---------------------------------------------------------------------------
// k3: Newton-Schulz inverse sqrt of 64x64 cov; single block.
// Y,Z,T in padded LDS; in-place Y=Y@T (per-row cache) and Z=T@Z (per-col
// cache). Also emits bias = deconv @ mean so k4 can fold mean subtraction.
// ---------------------------------------------------------------------------
__global__ void __launch_bounds__(256)
k3_isqrt(const float* __restrict__ cov, const float* __restrict__ meanv,
         float* __restrict__ deconv, float* __restrict__ bias) {
  __shared__ float Y[NCH * 65];
  __shared__ float Z[NCH * 65];
  __shared__ float T[NCH * 65];
  __shared__ float red[256];
  const int tid = threadIdx.x;

  float sq = 0.f;
  for (int i = tid; i < NCH * NCH; i += 256) {
    float v = cov[i];
    Y[(i >> 6) * 65 + (i & 63)] = v;
    sq += v * v;
  }
  red[tid] = sq;
  __syncthreads();
  for (int s = 128; s > 0; s >>= 1) {
    if (tid < s) red[tid] += red[tid + s];
    __syncthreads();
  }
  const float normA = sqrtf(red[0]);
  const float rn    = 1.0f / normA;
  __syncthreads();

  for (int i = tid; i < NCH * NCH; i += 256) {
    int r = i >> 6, c = i & 63;
    Y[r * 65 + c] *= rn;
    Z[r * 65 + c] = (r == c) ? 1.f : 0.f;
  }
  __syncthreads();

  for (int it = 0; it < 5; ++it) {
    if (tid < NCH) {                     // T = 1.5 I - 0.5 * Z@Y
      float rc[64];
      #pragma unroll
      for (int k = 0; k < 64; ++k) rc[k] = Z[tid * 65 + k];
      for (int c = 0; c < 64; ++c) {
        float s = 0.f;
        #pragma unroll
        for (int k = 0; k < 64; ++k) s += rc[k] * Y[k * 65 + c];
        T[tid * 65 + c] = ((tid == c) ? 1.5f : 0.f) - 0.5f * s;
      }
    }
    __syncthreads();
    if (tid < NCH) {                     // Y = Y @ T (row r depends on Y row r only)
      float rc[64];
      #pragma unroll
      for (int k = 0; k < 64; ++k) rc[k] = Y[tid * 65 + k];
      for (int c = 0; c < 64; ++c) {
        float s = 0.f;
        #pragma unroll
        for (int k = 0; k < 64; ++k) s += rc[k] * T[k * 65 + c];
        Y[tid * 65 + c] = s;
      }
    }
    if (tid < NCH) {                     // Z = T @ Z (col c depends on Z col c only)
      float cc[64];
      #pragma unroll
      for (int k = 0; k < 64; ++k) cc[k] = Z[k * 65 + tid];
      for (int r = 0; r < 64; ++r) {
        float s = 0.f;
        #pragma unroll
        for (int k = 0; k < 64; ++k) s += T[r * 65 + k] * cc[k];
        Z[r * 65 + tid] = s;
      }
    }
    __syncthreads();
  }

  const float rsn = rsqrtf(normA);
  for (int i = tid; i < NCH * NCH; i += 256)
    deconv[i] = Z[(i >> 6) * 65 + (i & 63)] * rsn;
  if (tid < NCH) {
    float s = 0.f;
    #pragma unroll
    for (int k = 0; k < 64; ++k) s += Z[tid * 65 + k] * rsn * meanv[k];
    bias[tid] = s;
  }
}

// ---------------------------------------------------------------------------
// k4: out = deconv @ x1 - bias, streamed with WMMA f32 16x16x4.
// Each wave owns 2 groups of 16 contiguous columns (32 cols, never straddles
// an H*W plane since 65536 % 32 == 0); 4 M-tiles x 16 K-steps per group.
// ---------------------------------------------------------------------------
__global__ void __launch_bounds__(256)
k4_apply(const float* __restrict__ x, const float* __restrict__ deconv,
         const float* __restrict__ bias, float* __restrict__ out) {
  __shared__ float sD[NCH * 65];
  __shared__ float sB[NCH];
  const int tid = threadIdx.x;
  for (int i = tid; i < NCH * NCH; i += 256)
    sD[(i >> 6) * 65 + (i & 63)] = deconv[i];
  if (tid < NCH) sB[tid] = bias[tid];
  __syncthreads();

  const int lane = tid & 31;
  const int wave = tid >> 5;
  const int col  = lane & 15;
  const int koff = (lane < 16) ? 0 : 2;
  const int gw   = blockIdx.x * 8 + wave;
  const int nw   = gridDim.x * 8;

  for (int p = gw; p < TOTJ / 32; p += nw) {
    const int j0 = p << 5;
    const int n  = j0 >> 16;
    const int hw = j0 & (HWSZ - 1);
    const float* xb = x + ((n * NCH) << 16) + hw + col;

    v8f acc0[4] = {};
    v8f acc1[4] = {};
    #pragma unroll
    for (int kc = 0; kc < 16; ++kc) {
      const int    k0 = kc * 4 + koff;
      const float* pk = xb + (k0 << 16);
      v2f b0, b1;
      b0[0] = pk[0];   b0[1] = pk[65536];        // cols j0 .. j0+15
      b1[0] = pk[16];  b1[1] = pk[65536 + 16];   // cols j0+16 .. j0+31
      #pragma unroll
      for (int mt = 0; mt < 4; ++mt) {
        const int m = mt * 16 + col;
        v2f a;
        a[0] = sD[m * 65 + k0];
        a[1] = sD[m * 65 + k0 + 1];
        acc0[mt] = __builtin_amdgcn_wmma_f32_16x16x4_f32(false, a, false, b0,
                                                         (short)0, acc0[mt], false, false);
        acc1[mt] = __builtin_amdgcn_wmma_f32_16x16x4_f32(false, a, false, b1,
                                                         (short)0, acc1[mt], false, false);
      }
    }

    float* ob = out + ((n * NCH) << 16) + hw + col;
    #pragma unroll
    for (int mt = 0; mt < 4; ++mt) {
      #pragma unroll
      for (int r = 0; r < 8; ++r) {
        const int   m  = mt * 16 + r + ((lane < 16) ? 0 : 8);
        const float bv = sB[m];
        ob[m << 16]        = acc0[mt][r] - bv;
        ob[(m << 16) + 16] = acc1[mt][r] - bv;
      }
    }
  }
}

// ---------------------------------------------------------------------------
extern "C" void kernel_launch(void* const* d_in, const int* in_sizes, int n_in,
                              void* d_out, int out_size, void* d_ws, size_t ws_size,
                              hipStream_t stream) {
  (void)in_sizes; (void)n_in; (void)out_size;
  const float* x   = (const float*)d_in[0];
  float*       out = (float*)d_out;
  float*       ws  = (float*)d_ws;

  // workspace layout (floats)
  float* cov     = ws;            // 4096
  float* meanv   = ws + 4096;     // 64
  float* deconv  = ws + 4160;     // 4096
  float* bias    = ws + 8256;     // 64
  float* covPart = ws + 8320;     // NB1 * 4096 (+ NB1 * 64 mean partials)

  // deterministic partial-block count, clamped to available scratch
  size_t availf = ws_size / sizeof(float);
  int    NB1    = 256;
  if (availf > 8320) {
    size_t per  = (size_t)(NCH * NCH + NCH);             // 4160 floats / block
    size_t maxb = (availf - 8320) / per;
    if ((size_t)NB1 > maxb) NB1 = (int)maxb;
  } else {
    NB1 = 1;
  }
  if (NB1 < 1) NB1 = 1;
  float* meanPart = covPart + (size_t)NB1 * (NCH * NCH);

  hipLaunchKernelGGL(k1_cov_partial, dim3(NB1), dim3(512), 0, stream, x, covPart, meanPart);
  hipLaunchKernelGGL(k2_finalize,    dim3(1),   dim3(256), 0, stream, covPart, meanPart, cov, meanv, NB1);
  hipLaunchKernelGGL(k3_isqrt,       dim3(1),   dim3(256), 0, stream, cov, meanv, deconv, bias);
  hipLaunchKernelGGL(k4_apply,       dim3(4096), dim3(256), 0, stream, x, deconv, bias, out);
}